// PrunedRnntLoss_56281251446976
// MI455X (gfx1250) — compile-verified
//
#include <hip/hip_runtime.h>

// Pruned RNN-T loss, MI455X (gfx1250, wave32).
// Phase 1: memory-bound logsumexp over V=1024 (168 MB streamed once -> ~7us floor @23.3TB/s),
//          cross-lane sum reduced through v_wmma_f32_16x16x4_f32 (exact fp32 ones-trick).
// Phase 2: anti-diagonal DP per batch; px/py tiles staged global->LDS with the
//          Tensor Data Mover (tensor_load_to_lds + s_wait_tensorcnt), DP state in LDS.
// Phase 3: deterministic scalar finalize.

#define NEGV (-1e30f)

constexpr int Bb = 16;
constexpr int Tt = 512;
constexpr int Ss = 256;
constexpr int Rr = 5;     // S_RANGE
constexpr int Vv = 1024;

typedef __attribute__((ext_vector_type(2))) float        v2f;
typedef __attribute__((ext_vector_type(8))) float        v8f;
typedef __attribute__((ext_vector_type(4))) unsigned int v4u;
typedef __attribute__((ext_vector_type(8))) int          v8i;
typedef __attribute__((ext_vector_type(4))) int          v4i;

__device__ __forceinline__ float wave_max32(float m) {
#pragma unroll
  for (int off = 16; off > 0; off >>= 1)
    m = fmaxf(m, __shfl_xor(m, off, 32));
  return m;
}

// Exact fp32 cross-lane sum of one value per lane (wave32).
// B(4x16) carries partials: b.x -> rows K=0 (lanes 0-15) / K=2 (lanes 16-31), b.y = 0.
// A(16x4) = ones => D[m][n] = p[n] + p[n+16]; D VGPR0 lane l = p[l&15]+p[(l&15)+16];
// 4 xor-shuffles finish the 16-way sum.
__device__ __forceinline__ float wave_sum32_wmma(float p) {
  v2f a; a.x = 1.0f; a.y = 1.0f;
  v2f b; b.x = p;    b.y = 0.0f;
  v8f c = {};
  c = __builtin_amdgcn_wmma_f32_16x16x4_f32(false, a, false, b, (short)0, c,
                                            false, false);
  float t = c[0];
#pragma unroll
  for (int off = 8; off > 0; off >>= 1)
    t += __shfl_xor(t, off, 32);
  return t;
}

// ---- Tensor Data Mover: 1-row fp32 tile, global -> LDS (D# per ISA ch.8) ----
// This toolchain exposes the 6-arg builtin:
//   (uint32x4 g0, int32x8 g1, int32x4 g2, int32x4 g3, int32x8 ext, i32 cpol)
__device__ __forceinline__ void tdm_load_row_f32(unsigned lds_byte_off,
                                                 const float* gptr,
                                                 unsigned nelem) {
  const unsigned long long ga = (unsigned long long)gptr;
  v4u g0;
  g0[0] = 1u;                                          // count=1, user, no gather
  g0[1] = lds_byte_off;                                // lds_addr (bytes)
  g0[2] = (unsigned)(ga & 0xFFFFFFFFull);              // global_addr[31:0]
  g0[3] = (unsigned)((ga >> 32) & 0x01FFFFFFull)       // global_addr[56:32]
        | (2u << 30);                                  // type = 2 ("image")
  v8i g1;
  g1[0] = (int)(2u << 16);                             // wg_mask=0, data_size=2 (4B)
  g1[1] = (int)((nelem & 0xFFFFu) << 16);              // tensor_dim0[15:0]
  g1[2] = (int)((nelem >> 16) & 0xFFFFu)               // tensor_dim0[31:16]
        | (1 << 16);                                   // tensor_dim1 = 1
  g1[3] = (int)((nelem & 0xFFFFu) << 16);              // tile_dim0 = nelem
  g1[4] = 1;                                           // tile_dim1=1, tile_dim2=0
  g1[5] = (int)nelem;                                  // tensor_dim0_stride[31:0]
  g1[6] = 0;                                           // stride0[47:32], stride1[15:0]
  g1[7] = 0;                                           // stride1[47:16]
  v4i g2  = {0, 0, 0, 0};                              // dims >= 2 unused
  v4i g3  = {0, 0, 0, 0};
  v8i ext = {0, 0, 0, 0, 0, 0, 0, 0};                  // extended words unused (flat copy)
  __builtin_amdgcn_tensor_load_to_lds(g0, g1, g2, g3, ext, /*cpol=*/0);
}

// One wave per (b,t,r) row: stream 1024 fp32, logsumexp, emit px/py compact.
__global__ void lse_gather_kernel(const float* __restrict__ logits,
                                  const int* __restrict__ targets,
                                  const int* __restrict__ ranges,
                                  float* __restrict__ px_c,
                                  float* __restrict__ py_c) {
  const int wave = threadIdx.x >> 5;
  const int lane = threadIdx.x & 31;
  const int row  = blockIdx.x * 8 + wave;              // < B*T*R = 40960
  const float*  base = logits + (size_t)row * Vv;
  const float4* b4   = (const float4*)base;            // 256 float4 per row

  float4 x[8];
#pragma unroll
  for (int j = 0; j < 8; ++j)
    x[j] = b4[j * 32 + lane];                          // coalesced 512B/instr per wave
  __builtin_prefetch(base + Vv, 0, 1);                 // global_prefetch_b8: next row

  float m = x[0].x;
#pragma unroll
  for (int j = 0; j < 8; ++j)
    m = fmaxf(m, fmaxf(fmaxf(x[j].x, x[j].y), fmaxf(x[j].z, x[j].w)));
  m = wave_max32(m);

  float p = 0.0f;
#pragma unroll
  for (int j = 0; j < 8; ++j)
    p += __expf(x[j].x - m) + __expf(x[j].y - m) +
         __expf(x[j].z - m) + __expf(x[j].w - m);

  const float tot  = wave_sum32_wmma(p);               // EXEC all-ones here
  const float norm = m + __logf(tot);

  if (lane == 0) {
    const int b    = row / (Tt * Rr);
    const int sidx = ranges[row];                      // absolute symbol position
    const int symv = (sidx < Ss) ? targets[b * Ss + sidx] : 0;  // sym[S]=TERM=0
    px_c[row] = base[symv] - norm;                     // L0/L2-hot re-read
    py_c[row] = base[0]    - norm;                     // blank logit
  }
}

// Single struct => static LDS layout starts at byte 0; member offsets are D# lds_addr.
struct AlphaSmem {
  float px[Tt * Rr];          // byte offset 0
  float py[Tt * Rr];          // byte offset 10240
  int   start[Tt];
  float pbuf[2][Ss + 8];
};
constexpr unsigned kPxOff = 0u;
constexpr unsigned kPyOff = (unsigned)(Tt * Rr) * 4u;

// One workgroup per batch: 257-state wavefront over 768 anti-diagonals.
__global__ void rnnt_alpha_kernel(const float* __restrict__ px_c,
                                  const float* __restrict__ py_c,
                                  const int* __restrict__ ranges,
                                  const int* __restrict__ boundary,
                                  float* __restrict__ mi) {
  __shared__ AlphaSmem sm;

  const int b    = blockIdx.x;
  const int tid  = threadIdx.x;
  const int nthr = blockDim.x;

  // TDM: wave 0 (uniform branch) issues both tile descriptors, drains TENSORcnt.
  if (tid < 32) {
    tdm_load_row_f32(kPxOff, px_c + (size_t)b * Tt * Rr, (unsigned)(Tt * Rr));
    tdm_load_row_f32(kPyOff, py_c + (size_t)b * Tt * Rr, (unsigned)(Tt * Rr));
    __builtin_amdgcn_s_wait_tensorcnt(0);
  }
  for (int t = tid; t < Tt; t += nthr)
    sm.start[t] = ranges[((size_t)b * Tt + t) * Rr];   // ranges[b,t,0]

  const int s_end = boundary[b * 4 + 2];
  const int t_end = boundary[b * 4 + 3];
  const int s = tid;
  if (s <= Ss) sm.pbuf[0][s] = (s == 0) ? 0.0f : NEGV;
  if (s == 0 && (s_end + t_end) == 0) mi[b] = 0.0f;    // degenerate boundary
  __syncthreads();                                     // publishes TDM tiles too

  const int dmax = Ss + Tt;                            // 768 diagonals
  for (int d = 1; d <= dmax; ++d) {
    const float* pp = sm.pbuf[(d - 1) & 1];
    float*       pn = sm.pbuf[d & 1];
    if (s <= Ss) {
      const int t = d - s;
      float pv = NEGV;
      if (t >= 0 && t <= Tt) {
        float term1 = NEGV, term2 = NEGV;
        if (s >= 1 && t < Tt) {                        // emit symbol s at time t
          const int r = (s - 1) - sm.start[t];
          if ((unsigned)r < (unsigned)Rr) term1 = pp[s - 1] + sm.px[t * Rr + r];
        }
        if (t >= 1) {                                  // blank at time t-1
          const int r = s - sm.start[t - 1];
          if ((unsigned)r < (unsigned)Rr) term2 = pp[s] + sm.py[(t - 1) * Rr + r];
        }
        const float mx = fmaxf(term1, term2);
        pv = mx + __logf(__expf(term1 - mx) + __expf(term2 - mx));
      }
      pn[s] = pv;
      if (d == s_end + t_end && s == s_end) mi[b] = pv;
    }
    __syncthreads();
  }
}

__global__ void finalize_kernel(const float* __restrict__ mi,
                                float* __restrict__ out) {
  if (threadIdx.x == 0) {
    float acc = 0.0f;
    for (int b = 0; b < Bb; ++b) acc += mi[b];         // deterministic order
    out[0] = -acc / (float)Bb;
  }
}

extern "C" void kernel_launch(void* const* d_in, const int* in_sizes, int n_in,
                              void* d_out, int out_size, void* d_ws, size_t ws_size,
                              hipStream_t stream) {
  const float* logits   = (const float*)d_in[0];
  const int*   targets  = (const int*)  d_in[1];
  // d_in[2]=logit_lengths, d_in[3]=target_lengths (unused; boundary carries both)
  const int*   boundary = (const int*)  d_in[4];
  const int*   ranges   = (const int*)  d_in[5];

  float* px_c = (float*)d_ws;                       // B*T*R fp32 = 160 KB
  float* py_c = px_c + (size_t)Bb * Tt * Rr;        // 160 KB
  float* mi   = py_c + (size_t)Bb * Tt * Rr;        // 16 fp32

  const int rows = Bb * Tt * Rr;                    // 40960, one wave each
  lse_gather_kernel<<<rows / 8, 256, 0, stream>>>(logits, targets, ranges,
                                                  px_c, py_c);
  rnnt_alpha_kernel<<<Bb, 288, 0, stream>>>(px_c, py_c, ranges, boundary, mi);
  finalize_kernel<<<1, 32, 0, stream>>>(mi, (float*)d_out);
}